// QueryAndGroup_22505628631263
// MI455X (gfx1250) — compile-verified
//
#include <hip/hip_runtime.h>

#define NPTS_ALL   8192
#define NPTS_VOX   4096
#define NPOINTS    1024
#define NSAMPLE    32
#define NBATCH     4
#define RAD2       0.04f

// d_out layout (floats), outputs concatenated in return order:
// new_xyz (4,1024,3) | new_normals (4,1024,3) | new_features (4,3,1024,32) | new_features_normals (4,6,1024,32)
#define OXYZ   0
#define ONRM   (NBATCH * NPOINTS * 3)                       // 12288
#define OFEAT  (2 * NBATCH * NPOINTS * 3)                   // 24576
#define OFEATN (OFEAT + NBATCH * 3 * NPOINTS * NSAMPLE)     // 417792

typedef float v2f __attribute__((ext_vector_type(2)));
typedef float v8f __attribute__((ext_vector_type(8)));

// ---------------------------------------------------------------------------
// Kernel 1: farthest point sampling (serial scan) + gather new_xyz/new_normals.
// One workgroup per batch; xyz_voxel resident in LDS; dists in VGPRs.
// ---------------------------------------------------------------------------
__global__ __launch_bounds__(256) void fps_gather_kernel(
    const float* __restrict__ xyz_voxel,
    const float* __restrict__ normals_voxel,
    float* __restrict__ out)
{
    const int b    = blockIdx.x;
    const int tid  = threadIdx.x;
    const int lane = tid & 31;   // wave32
    const int wave = tid >> 5;

    __shared__ float sx[NPTS_VOX], sy[NPTS_VOX], sz[NPTS_VOX];
    __shared__ int   chosen[NPOINTS];
    __shared__ float wbd[8];
    __shared__ int   wbi[8];
    __shared__ int   bnext;

    const float* xv = xyz_voxel + (size_t)b * NPTS_VOX * 3;
    for (int p = tid; p < NPTS_VOX; p += 256) {
        sx[p] = xv[p * 3 + 0];
        sy[p] = xv[p * 3 + 1];
        sz[p] = xv[p * 3 + 2];
    }
    if (tid == 0) chosen[0] = 0;
    __syncthreads();

    float dists[16];
#pragma unroll
    for (int k = 0; k < 16; ++k) dists[k] = 1.0e10f;

    float lx = sx[0], ly = sy[0], lz = sz[0];

    for (int it = 1; it < NPOINTS; ++it) {
        float bd = -1.0f;
        int   bi = 0;
#pragma unroll
        for (int k = 0; k < 16; ++k) {
            int   p  = k * 256 + tid;            // strided: bank-conflict free
            float dx = sx[p] - lx;
            float dy = sy[p] - ly;
            float dz = sz[p] - lz;
            float d  = dx * dx + dy * dy + dz * dz;
            float dm = fminf(dists[k], d);
            dists[k] = dm;
            if (dm > bd) { bd = dm; bi = p; }    // strict > keeps lowest p (argmax-first)
        }
        // intra-wave butterfly reduce (wave32), tie-break to lowest index
#pragma unroll
        for (int m = 16; m >= 1; m >>= 1) {
            float od = __shfl_xor(bd, m, 32);
            int   oi = __shfl_xor(bi, m, 32);
            if (od > bd || (od == bd && oi < bi)) { bd = od; bi = oi; }
        }
        if (lane == 0) { wbd[wave] = bd; wbi[wave] = bi; }
        __syncthreads();
        if (tid == 0) {
            float bb = wbd[0];
            int   ii = wbi[0];
#pragma unroll
            for (int w = 1; w < 8; ++w) {
                if (wbd[w] > bb || (wbd[w] == bb && wbi[w] < ii)) { bb = wbd[w]; ii = wbi[w]; }
            }
            chosen[it] = ii;
            bnext      = ii;
        }
        __syncthreads();
        int last = bnext;
        lx = sx[last]; ly = sy[last]; lz = sz[last];
    }
    __syncthreads();

    // gather selected points + normals to d_out
    const float* nv = normals_voxel + (size_t)b * NPTS_VOX * 3;
    for (int i = tid; i < NPOINTS; i += 256) {
        int    c  = chosen[i];
        size_t o  = (size_t)OXYZ + (size_t)b * NPOINTS * 3 + (size_t)i * 3;
        out[o + 0] = sx[c];
        out[o + 1] = sy[c];
        out[o + 2] = sz[c];
        size_t on = (size_t)ONRM + (size_t)b * NPOINTS * 3 + (size_t)i * 3;
        out[on + 0] = nv[c * 3 + 0];
        out[on + 1] = nv[c * 3 + 1];
        out[on + 2] = nv[c * 3 + 2];
    }
}

// ---------------------------------------------------------------------------
// Kernel 2: ball query via V_WMMA_F32_16X16X4_F32 + ordered selection + grouping.
// Block = 256 threads (8 waves) handles 16 queries of one batch.
// A row (16x4) = (qx,qy,qz,1); B col (4x16) = (-2px,-2py,-2pz,|p|^2);
// C preloaded with |q|^2 per row  =>  D = d^2 directly from one WMMA.
// ---------------------------------------------------------------------------
__global__ __launch_bounds__(256) void ball_group_kernel(
    const float* __restrict__ xyz_all,
    const float* __restrict__ normals,
    float* __restrict__ out)
{
    const int  tid  = threadIdx.x;
    const int  b    = blockIdx.x >> 6;          // 64 blocks per batch
    const int  q0   = (blockIdx.x & 63) * 16;
    const int  lane = tid & 31;
    const int  wave = tid >> 5;
    const int  r    = lane & 15;
    const bool hi   = lane >= 16;

    __shared__ float sq[16][4];                 // qx,qy,qz,|q|^2
    __shared__ float sd2[16 * 516];             // 16 rows, stride 516 (516 % 64 != 0)
    __shared__ int   sidx[16][NSAMPLE];
    __shared__ int   scnt[16];
    __shared__ int   sdone;

    const float* qptr = out + (size_t)OXYZ + (size_t)b * NPOINTS * 3;
    if (tid < 16) {
        float qx = qptr[(q0 + tid) * 3 + 0];
        float qy = qptr[(q0 + tid) * 3 + 1];
        float qz = qptr[(q0 + tid) * 3 + 2];
        sq[tid][0] = qx; sq[tid][1] = qy; sq[tid][2] = qz;
        sq[tid][3] = qx * qx + qy * qy + qz * qz;
        scnt[tid]  = 0;
    }
    if (tid == 0) sdone = 0;
    __syncthreads();

    // A-matrix 16x4 f32 layout: lanes 0-15 hold K=0 (v0), K=1 (v1); lanes 16-31 hold K=2, K=3
    v2f A;
    A[0] = hi ? sq[r][2] : sq[r][0];
    A[1] = hi ? 1.0f     : sq[r][1];

    // C-matrix bias: VGPR g holds rows (g, g+8); preload |q|^2 per row (loop-invariant)
    const int rbase = hi ? 8 : 0;
    v8f cinit;
#pragma unroll
    for (int g = 0; g < 8; ++g) cinit[g] = sq[rbase + g][3];

    const float* P = xyz_all + (size_t)b * NPTS_ALL * 3;

    for (int chunk = 0; chunk < NPTS_ALL / 512; ++chunk) {
        if (sdone) break;                        // block-uniform
        // Phase A: 8 waves x 4 tiles of 16 cols = 512 d^2 values
#pragma unroll
        for (int t = 0; t < 4; ++t) {
            int   col = wave * 64 + t * 16 + r;  // col within chunk [0,512)
            int   pi  = chunk * 512 + col;       // point index within batch
            float px = P[pi * 3 + 0];
            float py = P[pi * 3 + 1];
            float pz = P[pi * 3 + 2];
            float pn = px * px + py * py + pz * pz;
            v2f B;
            B[0] = hi ? (-2.0f * pz) : (-2.0f * px);
            B[1] = hi ? pn           : (-2.0f * py);
            v8f acc = __builtin_amdgcn_wmma_f32_16x16x4_f32(
                /*neg_a=*/false, A, /*neg_b=*/false, B,
                /*c_mod=*/(short)0, cinit, /*reuse_a=*/false, /*reuse_b=*/false);
#pragma unroll
            for (int g = 0; g < 8; ++g) {
                sd2[(rbase + g) * 516 + col] = acc[g];
            }
        }
        __syncthreads();
        // Phase B: ordered append of first indices with d^2 < r^2
        if (tid < 16) {
            int cnt = scnt[tid];
            for (int j = 0; j < 512; ++j) {
                if (cnt < NSAMPLE) {
                    float d2 = sd2[tid * 516 + j];
                    if (d2 < RAD2) sidx[tid][cnt++] = chunk * 512 + j;
                }
            }
            scnt[tid] = cnt;
        }
        __syncthreads();
        if (tid == 0) {
            int all = 1;
#pragma unroll
            for (int q = 0; q < 16; ++q) all &= (scnt[q] >= NSAMPLE) ? 1 : 0;
            sdone = all;
        }
        __syncthreads();
    }

    // Fill remainder with first hit (or 0 if no hits) per reference semantics
    if (tid < 16) {
        int cnt = scnt[tid];
        int f   = (cnt > 0) ? sidx[tid][0] : 0;
        for (int s = cnt; s < NSAMPLE; ++s) sidx[tid][s] = f;
    }
    __syncthreads();

    // Write grouped outputs: 16 queries x 32 samples over 256 threads
    const float* NRM = normals + (size_t)b * NPTS_ALL * 3;
    for (int pair = tid; pair < 16 * NSAMPLE; pair += 256) {
        int q = pair >> 5, s = pair & 31;
        int Q = q0 + q;
        int i = sidx[q][s];
        float r0 = P[i * 3 + 0] - sq[q][0];
        float r1 = P[i * 3 + 1] - sq[q][1];
        float r2 = P[i * 3 + 2] - sq[q][2];
        size_t fo = (size_t)OFEAT + (size_t)b * 3 * NPOINTS * NSAMPLE
                  + (size_t)Q * NSAMPLE + s;
        out[fo]                         = r0;
        out[fo + 1 * NPOINTS * NSAMPLE] = r1;
        out[fo + 2 * NPOINTS * NSAMPLE] = r2;
        size_t go = (size_t)OFEATN + (size_t)b * 6 * NPOINTS * NSAMPLE
                  + (size_t)Q * NSAMPLE + s;
        out[go]                         = r0;
        out[go + 1 * NPOINTS * NSAMPLE] = r1;
        out[go + 2 * NPOINTS * NSAMPLE] = r2;
        out[go + 3 * NPOINTS * NSAMPLE] = NRM[i * 3 + 0];
        out[go + 4 * NPOINTS * NSAMPLE] = NRM[i * 3 + 1];
        out[go + 5 * NPOINTS * NSAMPLE] = NRM[i * 3 + 2];
    }
}

extern "C" void kernel_launch(void* const* d_in, const int* in_sizes, int n_in,
                              void* d_out, int out_size, void* d_ws, size_t ws_size,
                              hipStream_t stream)
{
    (void)in_sizes; (void)n_in; (void)out_size; (void)d_ws; (void)ws_size;
    const float* xyz_all       = (const float*)d_in[0];
    const float* normals       = (const float*)d_in[1];
    const float* xyz_voxel     = (const float*)d_in[2];
    const float* normals_voxel = (const float*)d_in[3];
    float* out = (float*)d_out;

    fps_gather_kernel<<<NBATCH, 256, 0, stream>>>(xyz_voxel, normals_voxel, out);
    ball_group_kernel<<<NBATCH * (NPOINTS / 16), 256, 0, stream>>>(xyz_all, normals, out);
}